// CustomConv2d_19146964206453
// MI455X (gfx1250) — compile-verified
//
#include <hip/hip_runtime.h>
#include <stdint.h>

// Conv2d 3x3 VALID, NCHW fp32 -> implicit GEMM on CDNA5 WMMA (gfx1250).
// N=16, C_IN=32, H=W=512, C_OUT=32 -> out 16x32x510x510.
// Memory-bound (AI ~72 FLOP/B): fp32 end-to-end via V_WMMA_F32_16X16X4_F32.
// x tile staged once per block via global_load_async_to_lds_b32 into a
// ci-pair-interleaved LDS layout so every A/B fragment is a single aligned
// ds_load_b64 with an immediate offset (no address VALU, no pack movs).

typedef __attribute__((ext_vector_type(2))) float v2f;
typedef __attribute__((ext_vector_type(8))) float v8f;

#define CIN   32
#define COUT  32
#define HH    512
#define WW    512
#define KSZ   3
#define HOUT  510
#define WOUT  510

#define XROWS  10   // 8 output rows + 2 halo
#define XCOLSP 40   // 32 outputs + 2 halo, padded -> lane-half stride 400 pairs
                    // == 16 (mod 32): conflict-free b64 A reads
#define WCOP   48   // weight co stride: lane-half stride 48 pairs == 16 (mod 32)

// Block: 256 threads = 8 waves. Wave w -> output row ho_base+w, 32 wo, all co.
// Grid = n(16) * hoblk(64) * woseg(16) = 16384 blocks.
__launch_bounds__(256, 2)
__global__ void conv3x3_wmma_f32(const float* __restrict__ x,
                                 const float* __restrict__ wgt,
                                 const float* __restrict__ bias,
                                 float* __restrict__ out) {
  // x tile, ci-pair interleaved: [cp][lrow][col][j], ci = 2*cp + j  (50KB)
  __shared__ float xlds2[(CIN / 2) * XROWS * XCOLSP * 2];
  // weights, K-pair interleaved + padded co: [k/2][coP][2], k=(kh*3+kw)*32+ci
  __shared__ float wlds2[(KSZ * KSZ * CIN / 2) * WCOP * 2];   // 54KB
  // per-wave output transpose buffer (padded rows, 2 passes over co tiles)
  __shared__ float olds[8][16][33];

  const int tid    = threadIdx.x;
  const int wave   = tid >> 5;
  const int lane   = tid & 31;
  const int laneLo = lane & 15;
  const int laneHi = lane >> 4;   // 0 or 1

  const int bid  = blockIdx.x;
  const int wseg = bid & 15;          // 16 segments of 32 wo
  const int hblk = (bid >> 4) & 63;   // 64 blocks of 8 ho rows
  const int n    = bid >> 10;         // 0..15

  const int ho0 = hblk * 8;
  const int wo0 = wseg * 32;

  const size_t plane = (size_t)HH * WW;
  const float* xn    = x + (size_t)n * CIN * plane;

  // ---- Stage x tile: per-lane 4B async copies into interleaved layout. ----
  // i = ((cp*XROWS + lrow)*XCOLSP + col)*2 + j; consecutive lanes cover
  // 2 ci-planes x 16 consecutive cols -> two full 64B lines per instruction.
  // Edge rows/cols clamp the global address (dupes land in never-read slots).
  {
    const uint32_t xbase = (uint32_t)(uintptr_t)xlds2;
    for (int i = tid; i < (CIN / 2) * XROWS * XCOLSP * 2; i += 256) {
      const int j    = i & 1;
      const int t    = i >> 1;
      const int col  = t % XCOLSP;
      const int u    = t / XCOLSP;
      const int lrow = u % XROWS;
      const int cp   = u / XROWS;
      const int ci   = 2 * cp + j;
      int grow = ho0 + lrow; grow = grow > HH - 1 ? HH - 1 : grow;
      int gcol = wo0 + col;  gcol = gcol > WW - 1 ? WW - 1 : gcol;
      const float* gsrc = xn + (size_t)ci * plane + (size_t)grow * WW + gcol;
      const uint32_t ldsoff = xbase + 4u * (uint32_t)i;
      asm volatile("global_load_async_to_lds_b32 %0, %1, off"
                   :: "v"(ldsoff), "v"(gsrc) : "memory");
    }
  }

  // ---- Weights: OIHW -> [k/2][coP][2] pair-interleaved. ----
  for (int i = tid; i < KSZ * KSZ * CIN * COUT; i += 256) {
    const int k   = i >> 5;        // 0..287
    const int co  = i & 31;
    const int ci  = k & 31;
    const int khw = k >> 5;        // kh*3+kw
    const int kh  = khw / 3;
    const int kw  = khw % 3;
    wlds2[((k >> 1) * WCOP + co) * 2 + (k & 1)] =
        wgt[((co * CIN + ci) * KSZ + kh) * KSZ + kw];
  }

  asm volatile("s_wait_asynccnt 0" ::: "memory");  // this wave's async copies
  __syncthreads();                                  // all waves' writes visible

  const int ho = ho0 + wave;
  if (ho >= HOUT) return;   // wave-uniform: EXEC stays all-ones below

  // Accumulators: acc[mtile][cotile], each 16(M=wo) x 16(N=co) f32.
  v8f acc[2][2];
  acc[0][0] = (v8f){}; acc[0][1] = (v8f){};
  acc[1][0] = (v8f){}; acc[1][1] = (v8f){};

  // Per-lane fragment base pointers; all per-chunk offsets are compile-time
  // immediates (< 64KB -> DS 16-bit offset field).
  const float* pA = xlds2 + laneHi * (XROWS * XCOLSP * 2)   // +1 cp for hi half
                         + wave * (XCOLSP * 2) + laneLo * 2;
  const float* pB = wlds2 + laneHi * (WCOP * 2) + laneLo * 2;

#pragma unroll
  for (int kh = 0; kh < KSZ; ++kh) {
#pragma unroll
    for (int kw = 0; kw < KSZ; ++kw) {
#pragma unroll
      for (int c8 = 0; c8 < 8; ++c8) {
        // A 16x4 f32: VGPR0 = ci (K{0|2}), VGPR1 = ci+1 (K{1|3}) -> one b64.
        const int aoff = c8 * (2 * XROWS * XCOLSP)   // cp += 1 per c8
                       + kh * (XCOLSP * 2) + kw * 2;
        const v2f a0 = *(const v2f*)(pA + aoff);
        const v2f a1 = *(const v2f*)(pA + aoff + 32);   // mtile1: col +16

        // B 4x16: one aligned b64 per co tile.
        const int kp   = ((kh * 3 + kw) * 32 + 4 * c8) >> 1;
        const int boff = kp * (WCOP * 2);
        const v2f b0 = *(const v2f*)(pB + boff);
        const v2f b1 = *(const v2f*)(pB + boff + 32);   // co tile 1: co +16

        acc[0][0] = __builtin_amdgcn_wmma_f32_16x16x4_f32(
            false, a0, false, b0, (short)0, acc[0][0], false, false);
        acc[0][1] = __builtin_amdgcn_wmma_f32_16x16x4_f32(
            false, a0, false, b1, (short)0, acc[0][1], false, false);
        acc[1][0] = __builtin_amdgcn_wmma_f32_16x16x4_f32(
            false, a1, false, b0, (short)0, acc[1][0], false, false);
        acc[1][1] = __builtin_amdgcn_wmma_f32_16x16x4_f32(
            false, a1, false, b1, (short)0, acc[1][1], false, false);
      }
    }
  }

  // D layout: VGPR r -> M = r + 8*laneHi (+16*mtile), N(co) = laneLo (+16*ct).
  // Transpose via padded LDS so stores are 128B-contiguous per co row.
  float (*obuf)[33] = olds[wave];
  const size_t obase = (((size_t)n * COUT) * HOUT + ho) * WOUT;
  const int    wo    = wo0 + lane;
  const bool   ok    = wo < WOUT;

#pragma unroll
  for (int ct = 0; ct < 2; ++ct) {
#pragma unroll
    for (int r = 0; r < 8; ++r) {
      const int m0 = r + 8 * laneHi;
      obuf[laneLo][m0]      = acc[0][ct][r];
      obuf[laneLo][m0 + 16] = acc[1][ct][r];
    }
#pragma unroll
    for (int c = 0; c < 16; ++c) {
      const int co = ct * 16 + c;
      const float v = obuf[c][lane] + bias[co];   // bias: scalar load
      if (ok) out[obase + (size_t)co * (HOUT * WOUT) + wo] = v;
    }
  }
}

extern "C" void kernel_launch(void* const* d_in, const int* in_sizes, int n_in,
                              void* d_out, int out_size, void* d_ws, size_t ws_size,
                              hipStream_t stream) {
  (void)in_sizes; (void)n_in; (void)out_size; (void)d_ws; (void)ws_size;
  const float* x    = (const float*)d_in[0];
  const float* wgt  = (const float*)d_in[1];
  const float* bias = (const float*)d_in[2];
  float*       out  = (float*)d_out;

  dim3 grid(16 * 64 * 16);   // n * hoblk * woseg
  dim3 block(256);
  conv3x3_wmma_f32<<<grid, block, 0, stream>>>(x, wgt, bias, out);
}